// GatedKalmaNet_16690242912566
// MI455X (gfx1250) — compile-verified
//
#include <hip/hip_runtime.h>
#include <math.h>
#include <stdint.h>

// ---- problem dims ----
#define D_MODEL 2048
#define HDM 128
#define HQ 16
#define HK 8
#define CONVK 4
#define CHUNK 64
#define L_SEQ 1024
#define NC (L_SEQ / CHUNK)   // 16
#define NITER 30
#define RIDGE 0.02f
#define EPSV 1e-6f

typedef float v2f __attribute__((ext_vector_type(2)));
typedef float v8f __attribute__((ext_vector_type(8)));

// f32 WMMA: D(16x16,f32) = A(16x4,f32) x B(4x16,f32) + C
__device__ __forceinline__ v8f wmma_f32_k4(v2f a, v2f b, v8f c) {
  return __builtin_amdgcn_wmma_f32_16x16x4_f32(false, a, false, b, (short)0, c,
                                               false, false);
}

__device__ __forceinline__ float wave_reduce_add(float v) {
  for (int off = 16; off > 0; off >>= 1) v += __shfl_down(v, off, 32);
  return v;
}

// ---------------------------------------------------------------------------
// Generic f32 GEMM: C[M,N] = A[M,K] @ B[K,N], row-major. 64x64 tile / block,
// 256 threads = 8 waves, each wave owns two 16x16 WMMA tiles.
// Double-buffered GLOBAL_LOAD_ASYNC_TO_LDS_B128 staging (ASYNCcnt pipeline):
// tile t+1 copies in flight while WMMA consumes tile t from LDS.
// M,N multiples of 64; K multiple of 16.
// ---------------------------------------------------------------------------
__global__ void gemm_f32_wmma(const float* __restrict__ A,
                              const float* __restrict__ B,
                              float* __restrict__ C, int M, int N, int K) {
  // A pad stride 20 floats: rows stay 16B aligned for b128 async writes and
  // (20*r)%64 is distinct for r=0..15 -> conflict-free column fragment reads.
  __shared__ __align__(16) float As[2][64 * 20];
  __shared__ __align__(16) float Bs[2][16 * 64];
  const int tid = threadIdx.x;
  const int lane = tid & 31, wave = tid >> 5;
  const int half = lane >> 4, r = lane & 15;
  const int mblk = blockIdx.y * 64, nblk = blockIdx.x * 64;
  const int ti = wave >> 1, tj0 = (wave & 1) * 2;
  // per-thread async staging coords: 4 floats (16B) of As and of Bs each
  const int arow = tid >> 2, acol = (tid & 3) * 4;    // 64 x 16 tile
  const int brow = tid >> 4, bcol = (tid & 15) * 4;   // 16 x 64 tile
  auto stage = [&](int buf, int k0) {
    uint64_t ga =
        (uint64_t)(uintptr_t)(A + (size_t)(mblk + arow) * K + k0 + acol);
    uint32_t la = (uint32_t)(uintptr_t)&As[buf][arow * 20 + acol];
    asm volatile("global_load_async_to_lds_b128 %0, %1, off" ::"v"(la),
                 "v"(ga)
                 : "memory");
    uint64_t gb =
        (uint64_t)(uintptr_t)(B + (size_t)(k0 + brow) * N + nblk + bcol);
    uint32_t lb = (uint32_t)(uintptr_t)&Bs[buf][brow * 64 + bcol];
    asm volatile("global_load_async_to_lds_b128 %0, %1, off" ::"v"(lb),
                 "v"(gb)
                 : "memory");
  };
  v8f acc0 = {}, acc1 = {};
  const int nk = K / 16;
  stage(0, 0);
  for (int t = 0; t < nk; ++t) {
    const int buf = t & 1;
    if (t + 1 < nk) {
      stage(buf ^ 1, (t + 1) * 16);
      // 2 newest copies may stay in flight; the 2 for tile t must be done
      asm volatile("s_wait_asynccnt 0x2" ::: "memory");
    } else {
      asm volatile("s_wait_asynccnt 0x0" ::: "memory");
    }
    __syncthreads();
#pragma unroll
    for (int kk = 0; kk < 16; kk += 4) {
      v2f a = *(const v2f*)&As[buf][(ti * 16 + r) * 20 + kk + 2 * half];
      v2f b0, b1;
      b0.x = Bs[buf][(kk + 2 * half) * 64 + tj0 * 16 + r];
      b0.y = Bs[buf][(kk + 2 * half + 1) * 64 + tj0 * 16 + r];
      b1.x = Bs[buf][(kk + 2 * half) * 64 + (tj0 + 1) * 16 + r];
      b1.y = Bs[buf][(kk + 2 * half + 1) * 64 + (tj0 + 1) * 16 + r];
      acc0 = wmma_f32_k4(a, b0, acc0);
      acc1 = wmma_f32_k4(a, b1, acc1);
    }
    __syncthreads();  // safe to overwrite this buffer two tiles later
  }
#pragma unroll
  for (int i = 0; i < 8; ++i) {
    int m = mblk + ti * 16 + i + 8 * half;
    C[(size_t)m * N + nblk + tj0 * 16 + r] = acc0[i];
    C[(size_t)m * N + nblk + (tj0 + 1) * 16 + r] = acc1[i];
  }
}

// ---------------------------------------------------------------------------
// Fused gate projections (N=16 each) + nonlinearities. One block per token.
// ---------------------------------------------------------------------------
__global__ void gates_kernel(const float* __restrict__ x,
                             const float* __restrict__ Wa,
                             const float* __restrict__ Wb,
                             const float* __restrict__ Wal,
                             const float* __restrict__ A_log,
                             const float* __restrict__ dt_bias,
                             const float* __restrict__ bb,
                             const float* __restrict__ balpha,
                             float* __restrict__ glog, float* __restrict__ beta,
                             float* __restrict__ alpha) {
  const int l = blockIdx.x;
  __shared__ float xs[D_MODEL];
  for (int i = threadIdx.x; i < D_MODEL; i += blockDim.x)
    xs[i] = x[(size_t)l * D_MODEL + i];
  __syncthreads();
  const int wave = threadIdx.x >> 5, lane = threadIdx.x & 31;
  for (int h = wave; h < HQ; h += 4) {
    float sa = 0.f, sb = 0.f, sl = 0.f;
    for (int kk = lane; kk < D_MODEL; kk += 32) {
      float xv = xs[kk];
      sa += xv * Wa[kk * HQ + h];
      sb += xv * Wb[kk * HQ + h];
      sl += xv * Wal[kk * HQ + h];
    }
    sa = wave_reduce_add(sa);
    sb = wave_reduce_add(sb);
    sl = wave_reduce_add(sl);
    if (lane == 0) {
      float z = sa + dt_bias[h];
      float sp = (z > 20.f) ? z : log1pf(expf(z));
      glog[l * HQ + h] = -expf(A_log[h]) * sp;
      beta[l * HQ + h] = 1.f / (1.f + expf(-(sb + bb[h])));
      alpha[l * HQ + h] = 1.f / (1.f + expf(-(sl + balpha[h])));
    }
  }
}

// ---------------------------------------------------------------------------
// Causal depthwise conv (K=4) + SiLU (+ optional scale, q uses d^-0.5)
// ---------------------------------------------------------------------------
__global__ void conv_silu_kernel(const float* __restrict__ x,
                                 const float* __restrict__ w,
                                 float* __restrict__ y, int Cdim, float scale) {
  int idx = blockIdx.x * blockDim.x + threadIdx.x;
  if (idx >= L_SEQ * Cdim) return;
  int l = idx / Cdim, c = idx % Cdim;
  float s = 0.f;
#pragma unroll
  for (int j = 0; j < CONVK; ++j) {
    int ll = l - (CONVK - 1) + j;
    if (ll >= 0) s += x[(size_t)ll * Cdim + c] * w[c * CONVK + j];
  }
  y[idx] = s * (1.f / (1.f + expf(-s))) * scale;
}

// ---------------------------------------------------------------------------
// Per-(head, chunk) decay stats: inclusive cumsum of glog, w_end, dg.
// ---------------------------------------------------------------------------
__global__ void chunkstats_kernel(const float* __restrict__ glog,
                                  const float* __restrict__ beta,
                                  float* __restrict__ cs,
                                  float* __restrict__ wend,
                                  float* __restrict__ dg) {
  const int n = blockIdx.x % NC;
  __shared__ float s[CHUNK];
  const int c = threadIdx.x;  // 64 threads
  const int h = blockIdx.x / NC;
  s[c] = glog[(n * CHUNK + c) * HQ + h];
  __syncthreads();
  for (int off = 1; off < CHUNK; off <<= 1) {
    float v = (c >= off) ? s[c - off] : 0.f;
    __syncthreads();
    s[c] += v;
    __syncthreads();
  }
  float csv = s[c];
  float G = s[CHUNK - 1];
  cs[blockIdx.x * CHUNK + c] = csv;
  wend[blockIdx.x * CHUNK + c] = expf(G - csv) * beta[(n * CHUNK + c) * HQ + h];
  if (c == 0) dg[blockIdx.x] = expf(G);
}

// ---------------------------------------------------------------------------
// dH = K^T diag(w_end) K, dB = K^T diag(w_end) V  per (h, chunk). WMMA from LDS.
// ---------------------------------------------------------------------------
__global__ void chunk_outer_kernel(const float* __restrict__ kconv,
                                   const float* __restrict__ vconv,
                                   const float* __restrict__ wend,
                                   float* __restrict__ dH,
                                   float* __restrict__ dB) {
  extern __shared__ float sm0[];
  float* ks = sm0;                 // [64][128]
  float* wks = ks + CHUNK * HDM;   // [64][128]  w*k
  float* vs = wks + CHUNK * HDM;   // [64][128]
  const int b = blockIdx.x;
  const int h = b / NC, n = b % NC, hk = h >> 1;  // GQA: repeat (h g) order
  const int tid = threadIdx.x;
  for (int i = tid; i < CHUNK * HDM; i += 256) {
    int c = i >> 7, d = i & 127;
    float kv = kconv[(size_t)(n * CHUNK + c) * (HK * HDM) + hk * HDM + d];
    float vv = vconv[(size_t)(n * CHUNK + c) * (HK * HDM) + hk * HDM + d];
    ks[i] = kv;
    wks[i] = wend[b * CHUNK + c] * kv;
    vs[i] = vv;
  }
  __syncthreads();
  const int lane = tid & 31, wave = tid >> 5, half = lane >> 4, r = lane & 15;
  const size_t base = (size_t)b * HDM * HDM;
  for (int which = 0; which < 2; ++which) {
    const float* Bsrc = which ? vs : ks;
    float* dst = which ? dB : dH;
    v8f zero = {};
    v8f acc[8];
#pragma unroll
    for (int tj = 0; tj < 8; ++tj) acc[tj] = zero;
    for (int k0 = 0; k0 < CHUNK; k0 += 4) {
      v2f a;  // A[m][c] = wk[c][m]  (row-contiguous LDS read, no conflict)
      a.x = wks[(k0 + 2 * half) * HDM + wave * 16 + r];
      a.y = wks[(k0 + 2 * half + 1) * HDM + wave * 16 + r];
#pragma unroll
      for (int tj = 0; tj < 8; ++tj) {
        v2f bf;
        bf.x = Bsrc[(k0 + 2 * half) * HDM + tj * 16 + r];
        bf.y = Bsrc[(k0 + 2 * half + 1) * HDM + tj * 16 + r];
        acc[tj] = wmma_f32_k4(a, bf, acc[tj]);
      }
    }
#pragma unroll
    for (int tj = 0; tj < 8; ++tj)
#pragma unroll
      for (int i = 0; i < 8; ++i) {
        int m = wave * 16 + i + 8 * half;
        dst[base + (size_t)m * HDM + tj * 16 + r] = acc[tj][i];
      }
  }
}

// ---------------------------------------------------------------------------
// Sequential inter-chunk scan per head. Emits A = H_start + ridge*I and B0.
// ---------------------------------------------------------------------------
__global__ void scan_kernel(const float* __restrict__ dH,
                            const float* __restrict__ dB,
                            const float* __restrict__ dg,
                            float* __restrict__ Amat, float* __restrict__ B0) {
  const int h = blockIdx.x;
  const int t = threadIdx.x;  // 1024 threads, 16 elems each per matrix
  float hs[16], bs[16];
#pragma unroll
  for (int i = 0; i < 16; ++i) { hs[i] = 0.f; bs[i] = 0.f; }
  for (int n = 0; n < NC; ++n) {
    float g = dg[h * NC + n];
    size_t base = (size_t)(h * NC + n) * HDM * HDM;
#pragma unroll
    for (int i = 0; i < 16; ++i) {
      int e = t + i * 1024;
      int row = e >> 7, col = e & 127;
      Amat[base + e] = hs[i] + (row == col ? RIDGE : 0.f);
      B0[base + e] = bs[i];
      hs[i] = g * hs[i] + dH[base + e];
      bs[i] = g * bs[i] + dB[base + e];
    }
  }
}

__global__ void trace_kernel(const float* __restrict__ Amat,
                             float* __restrict__ lmax) {
  const int b = blockIdx.x;
  const int d = threadIdx.x;  // 128
  __shared__ float red[128];
  red[d] = Amat[((size_t)b * HDM + d) * HDM + d];
  __syncthreads();
  for (int off = 64; off > 0; off >>= 1) {
    if (d < off) red[d] += red[d + off];
    __syncthreads();
  }
  if (d == 0) lmax[b] = red[0];
}

// ---------------------------------------------------------------------------
// Fused 30-iteration Chebyshev solve, fully LDS-resident (258KB of 320KB WGP
// LDS). One block per (h, chunk). wave w owns output rows [16w, 16w+16).
// ---------------------------------------------------------------------------
__global__ void cheb_kernel(const float* __restrict__ Amat,
                            const float* __restrict__ B0,
                            const float* __restrict__ lmax,
                            float* __restrict__ X) {
  extern __shared__ float sm1[];
  float* As = sm1;              // 128 x 132 (pad: kills column-read conflicts)
  float* dv = As + 128 * 132;   // 128 x 128
  float* rr = dv + 128 * 128;
  float* xx = rr + 128 * 128;
  const int b = blockIdx.x;
  const size_t base = (size_t)b * HDM * HDM;
  const int tid = threadIdx.x;  // 256
  const float lmx = lmax[b];
  const float theta = (lmx + RIDGE) * 0.5f;
  const float delta = (lmx - RIDGE) * 0.5f;
  const float sigma1 = theta / delta;
  const float invtheta = 1.f / theta;
  for (int i = tid; i < HDM * HDM; i += 256) {
    int row = i >> 7, col = i & 127;
    As[row * 132 + col] = Amat[base + i];
    float bv = B0[base + i];
    rr[i] = bv;
    dv[i] = bv * invtheta;
    xx[i] = 0.f;
  }
  __syncthreads();
  const int lane = tid & 31, wave = tid >> 5, half = lane >> 4, r = lane & 15;
  float rho = 1.f / sigma1;
  for (int it = 0; it < NITER; ++it) {
    v8f zero = {};
    v8f acc[8];
#pragma unroll
    for (int tj = 0; tj < 8; ++tj) acc[tj] = zero;
    for (int k0 = 0; k0 < HDM; k0 += 4) {
      v2f a = *(const v2f*)&As[(wave * 16 + r) * 132 + k0 + 2 * half];
#pragma unroll
      for (int tj = 0; tj < 8; ++tj) {
        v2f bf;
        bf.x = dv[(k0 + 2 * half) * HDM + tj * 16 + r];
        bf.y = dv[(k0 + 2 * half + 1) * HDM + tj * 16 + r];
        acc[tj] = wmma_f32_k4(a, bf, acc[tj]);
      }
    }
    // x += dvec (old dvec; dvec not written until after the barrier)
    for (int i = tid; i < HDM * HDM; i += 256) xx[i] += dv[i];
    // r -= A@dvec : each wave writes its own rows
#pragma unroll
    for (int tj = 0; tj < 8; ++tj)
#pragma unroll
      for (int i = 0; i < 8; ++i) {
        int m = wave * 16 + i + 8 * half;
        rr[m * HDM + tj * 16 + r] -= acc[tj][i];
      }
    __syncthreads();
    float rho_n = 1.f / (2.f * sigma1 - rho);
    float c1 = rho_n * rho;
    float c2 = 2.f * rho_n / delta;
    for (int i = tid; i < HDM * HDM; i += 256) dv[i] = c1 * dv[i] + c2 * rr[i];
    rho = rho_n;
    __syncthreads();
  }
  for (int i = tid; i < HDM * HDM; i += 256) X[base + i] = xx[i];
}

// ---------------------------------------------------------------------------
// Readout per (h, chunk): S = (qK^T) .* mask .* exp(cs_c - cs_j) .* beta_j,
// O = diag(exp(cs)) (q @ X) + S @ V + alpha .* V
// ---------------------------------------------------------------------------
__global__ void readout_kernel(const float* __restrict__ qconv,
                               const float* __restrict__ kconv,
                               const float* __restrict__ vconv,
                               const float* __restrict__ Xbuf,
                               const float* __restrict__ cs,
                               const float* __restrict__ beta,
                               const float* __restrict__ alpha,
                               float* __restrict__ obuf) {
  extern __shared__ float sm2[];
  float* qs = sm2;                // 64 x 132
  float* ks = qs + 64 * 132;      // 64 x 132
  float* vs = ks + 64 * 132;      // 64 x 128
  float* Xs = vs + 64 * 128;      // 128 x 128
  float* Ss = Xs + 128 * 128;     // 64 x 68
  float* csb = Ss + 64 * 68;      // 64
  float* ecs = csb + 64;          // 64
  float* btb = ecs + 64;          // 64
  const int b = blockIdx.x;
  const int h = b / NC, n = b % NC, hk = h >> 1;
  const int tid = threadIdx.x;
  for (int i = tid; i < CHUNK * HDM; i += 256) {
    int c = i >> 7, d = i & 127;
    qs[c * 132 + d] = qconv[(size_t)(n * CHUNK + c) * (HQ * HDM) + h * HDM + d];
    ks[c * 132 + d] = kconv[(size_t)(n * CHUNK + c) * (HK * HDM) + hk * HDM + d];
    vs[c * 128 + d] = vconv[(size_t)(n * CHUNK + c) * (HK * HDM) + hk * HDM + d];
  }
  for (int i = tid; i < HDM * HDM; i += 256) Xs[i] = Xbuf[(size_t)b * HDM * HDM + i];
  for (int i = tid; i < CHUNK; i += 256) {
    float cv = cs[b * CHUNK + i];
    csb[i] = cv;
    ecs[i] = expf(cv);
    btb[i] = beta[(n * CHUNK + i) * HQ + h];
  }
  __syncthreads();
  const int lane = tid & 31, wave = tid >> 5, half = lane >> 4, r = lane & 15;
  // Phase 1: S = q @ k^T (64x64, K=128); gate in accumulator regs; stash to LDS
  {
    int ti = wave >> 1;
    int tjb = (wave & 1) * 2;
    for (int tjo = 0; tjo < 2; ++tjo) {
      int tj = tjb + tjo;
      v8f acc = {};
      for (int k0 = 0; k0 < HDM; k0 += 4) {
        // both operands are row-contiguous pairs -> single b64 LDS loads
        v2f a = *(const v2f*)&qs[(ti * 16 + r) * 132 + k0 + 2 * half];
        v2f bf = *(const v2f*)&ks[(tj * 16 + r) * 132 + k0 + 2 * half];
        acc = wmma_f32_k4(a, bf, acc);
      }
#pragma unroll
      for (int i = 0; i < 8; ++i) {
        int m = ti * 16 + i + 8 * half;  // row c
        int nn = tj * 16 + r;            // col j
        float g = (nn <= m) ? expf(csb[m] - csb[nn]) * btb[nn] : 0.f;
        Ss[m * 68 + nn] = acc[i] * g;
      }
    }
  }
  __syncthreads();
  // Phase 2: O = rowscale(q@X) then accumulate S@V via WMMA chaining on C
  {
    int ti = wave >> 1;
    int tjb = (wave & 1) * 4;
    for (int tjo = 0; tjo < 4; ++tjo) {
      int tj = tjb + tjo;
      v8f acc = {};
      for (int k0 = 0; k0 < HDM; k0 += 4) {
        v2f a = *(const v2f*)&qs[(ti * 16 + r) * 132 + k0 + 2 * half];
        v2f bf;
        bf.x = Xs[(k0 + 2 * half) * HDM + tj * 16 + r];
        bf.y = Xs[(k0 + 2 * half + 1) * HDM + tj * 16 + r];
        acc = wmma_f32_k4(a, bf, acc);
      }
#pragma unroll
      for (int i = 0; i < 8; ++i) acc[i] *= ecs[ti * 16 + i + 8 * half];
      for (int k0 = 0; k0 < CHUNK; k0 += 4) {
        v2f a = *(const v2f*)&Ss[(ti * 16 + r) * 68 + k0 + 2 * half];
        v2f bf;
        bf.x = vs[(k0 + 2 * half) * HDM + tj * 16 + r];
        bf.y = vs[(k0 + 2 * half + 1) * HDM + tj * 16 + r];
        acc = wmma_f32_k4(a, bf, acc);
      }
#pragma unroll
      for (int i = 0; i < 8; ++i) {
        int m = ti * 16 + i + 8 * half;  // c in chunk
        int l = n * CHUNK + m;
        int dcol = tj * 16 + r;
        float av = alpha[l * HQ + h];
        obuf[(size_t)l * (HQ * HDM) + h * HDM + dcol] =
            acc[i] + av * vs[m * HDM + dcol];
      }
    }
  }
}

// ---------------------------------------------------------------------------
// Gated RMSNorm (per token-head row of 128) in place.
// ---------------------------------------------------------------------------
__global__ void norm_gate_kernel(const float* __restrict__ gbuf,
                                 const float* __restrict__ norm_w,
                                 float* __restrict__ obuf) {
  const int bidx = blockIdx.x;  // l*HQ + h
  const int l = bidx / HQ, h = bidx % HQ;
  const int d = threadIdx.x;  // 128
  const size_t idx = (size_t)l * (HQ * HDM) + h * HDM + d;
  float o = obuf[idx];
  float w = wave_reduce_add(o * o);
  __shared__ float red[4];
  if ((threadIdx.x & 31) == 0) red[threadIdx.x >> 5] = w;
  __syncthreads();
  float tot = red[0] + red[1] + red[2] + red[3];
  float inv = rsqrtf(tot * (1.f / 128.f) + EPSV);
  float g = gbuf[idx];
  float sig = 1.f / (1.f + expf(-g));
  obuf[idx] = o * inv * norm_w[d] * (g * sig);
}

// ---------------------------------------------------------------------------
extern "C" void kernel_launch(void* const* d_in, const int* in_sizes, int n_in,
                              void* d_out, int out_size, void* d_ws,
                              size_t ws_size, hipStream_t stream) {
  (void)in_sizes; (void)n_in; (void)out_size; (void)ws_size;
  const float* x      = (const float*)d_in[0];
  const float* Wq     = (const float*)d_in[1];
  const float* Wk     = (const float*)d_in[2];
  const float* Wv     = (const float*)d_in[3];
  const float* conv_q = (const float*)d_in[4];
  const float* conv_k = (const float*)d_in[5];
  const float* conv_v = (const float*)d_in[6];
  const float* Wa     = (const float*)d_in[7];
  const float* A_log  = (const float*)d_in[8];
  const float* dt_b   = (const float*)d_in[9];
  const float* Wb     = (const float*)d_in[10];
  const float* bb     = (const float*)d_in[11];
  const float* Wal    = (const float*)d_in[12];
  const float* bal    = (const float*)d_in[13];
  const float* Wg     = (const float*)d_in[14];
  const float* normw  = (const float*)d_in[15];
  const float* Wo     = (const float*)d_in[16];
  float* out = (float*)d_out;

  float* ws = (float*)d_ws;
  size_t off = 0;
  auto alloc = [&](size_t n) { float* p = ws + off; off += n; return p; };
  float* qbuf  = alloc((size_t)L_SEQ * HQ * HDM);   // 1024x2048
  float* kbuf  = alloc((size_t)L_SEQ * HK * HDM);   // 1024x1024
  float* vbuf  = alloc((size_t)L_SEQ * HK * HDM);
  float* gbuf  = alloc((size_t)L_SEQ * HQ * HDM);
  float* qcv   = alloc((size_t)L_SEQ * HQ * HDM);
  float* kcv   = alloc((size_t)L_SEQ * HK * HDM);
  float* vcv   = alloc((size_t)L_SEQ * HK * HDM);
  float* glog  = alloc((size_t)L_SEQ * HQ);
  float* beta  = alloc((size_t)L_SEQ * HQ);
  float* alpha = alloc((size_t)L_SEQ * HQ);
  float* cs    = alloc((size_t)HQ * NC * CHUNK);
  float* wend  = alloc((size_t)HQ * NC * CHUNK);
  float* dg    = alloc((size_t)HQ * NC);
  float* dH    = alloc((size_t)HQ * NC * HDM * HDM);
  float* dB    = alloc((size_t)HQ * NC * HDM * HDM);
  float* Amat  = alloc((size_t)HQ * NC * HDM * HDM);
  float* B0    = alloc((size_t)HQ * NC * HDM * HDM);
  float* Xbuf  = alloc((size_t)HQ * NC * HDM * HDM);
  float* lmax  = alloc((size_t)HQ * NC);
  float* obuf  = alloc((size_t)L_SEQ * HQ * HDM);

  // 1) big projections (async-LDS double-buffered WMMA GEMM)
  gemm_f32_wmma<<<dim3(2048 / 64, 1024 / 64), 256, 0, stream>>>(x, Wq, qbuf, 1024, 2048, 2048);
  gemm_f32_wmma<<<dim3(1024 / 64, 1024 / 64), 256, 0, stream>>>(x, Wk, kbuf, 1024, 1024, 2048);
  gemm_f32_wmma<<<dim3(1024 / 64, 1024 / 64), 256, 0, stream>>>(x, Wv, vbuf, 1024, 1024, 2048);
  gemm_f32_wmma<<<dim3(2048 / 64, 1024 / 64), 256, 0, stream>>>(x, Wg, gbuf, 1024, 2048, 2048);
  // 2) gates
  gates_kernel<<<L_SEQ, 128, 0, stream>>>(x, Wa, Wb, Wal, A_log, dt_b, bb, bal,
                                          glog, beta, alpha);
  // 3) convs (+ q scale d^-0.5)
  conv_silu_kernel<<<(L_SEQ * HQ * HDM) / 256, 256, 0, stream>>>(qbuf, conv_q, qcv, HQ * HDM, 0.08838834764831845f);
  conv_silu_kernel<<<(L_SEQ * HK * HDM) / 256, 256, 0, stream>>>(kbuf, conv_k, kcv, HK * HDM, 1.f);
  conv_silu_kernel<<<(L_SEQ * HK * HDM) / 256, 256, 0, stream>>>(vbuf, conv_v, vcv, HK * HDM, 1.f);
  // 4) per-chunk decay stats
  chunkstats_kernel<<<HQ * NC, CHUNK, 0, stream>>>(glog, beta, cs, wend, dg);
  // 5) gram increments
  chunk_outer_kernel<<<HQ * NC, 256, 3 * CHUNK * HDM * sizeof(float), stream>>>(kcv, vcv, wend, dH, dB);
  // 6) inter-chunk scan (+ridge fold)
  scan_kernel<<<HQ, 1024, 0, stream>>>(dH, dB, dg, Amat, B0);
  // 7) lmax = trace(A)
  trace_kernel<<<HQ * NC, 128, 0, stream>>>(Amat, lmax);
  // 8) fused LDS-resident Chebyshev solve (258KB dynamic LDS of 320KB WGP LDS)
  {
    size_t lds = (size_t)(128 * 132 + 3 * 128 * 128) * sizeof(float);
    cheb_kernel<<<HQ * NC, 256, lds, stream>>>(Amat, B0, lmax, Xbuf);
  }
  // 9) readout (inter + intra + alpha residual)
  {
    size_t lds = (size_t)(64 * 132 * 2 + 64 * 128 + 128 * 128 + 64 * 68 + 3 * 64) * sizeof(float);
    readout_kernel<<<HQ * NC, 256, lds, stream>>>(qcv, kcv, vcv, Xbuf, cs, beta, alpha, obuf);
  }
  // 10) gated RMSNorm + output projection
  norm_gate_kernel<<<L_SEQ * HQ, 128, 0, stream>>>(gbuf, normw, obuf);
  gemm_f32_wmma<<<dim3(2048 / 64, 1024 / 64), 256, 0, stream>>>(obuf, Wo, out, 1024, 2048, 2048);
}